// DGLDigitCapsuleLayer_49289044689251
// MI455X (gfx1250) — compile-verified
//
#include <hip/hip_runtime.h>
#include <hip/hip_bf16.h>
#include <stdint.h>

// ---------------------------------------------------------------------------
// CapsNet dynamic routing for MI455X (gfx1250, wave32).
//  * u_hat einsum  -> V_WMMA_F32_16X16X4_F32 (full fp32 precision, K=8 = 2 steps)
//  * u_hat (188.7 MB) parked in workspace; fits MI455X 192 MB L2 -> routing
//    passes are L2-bandwidth streaming reductions.
// ---------------------------------------------------------------------------

typedef __attribute__((ext_vector_type(2))) float v2f;
typedef __attribute__((ext_vector_type(8))) float v8f;

#define IN_NUM   1152
#define OUT_NUM  10
#define OUT_DIM  16
#define IN_DIM   8
#define BATCH    256
#define NROUTE   3
#define ISPLIT   8           // parallel splits of the i-reduction in k_reduce_s

#define JBD      (OUT_NUM * BATCH * OUT_DIM)   // 40960 elements of s / v

// ---------------------------------------------------------------------------
// x[b][c][i]  ->  x_t[i][c][b]   (tiled LDS transpose; both sides coalesced)
// ---------------------------------------------------------------------------
__global__ void __launch_bounds__(256)
k_transpose_x(const float* __restrict__ x, float* __restrict__ xt) {
    __shared__ float tile[32][33];
    const int c  = blockIdx.z;            // 0..7
    const int i0 = blockIdx.x * 32;       // 36 tiles over i
    const int b0 = blockIdx.y * 32;       // 8 tiles over b
    const int tx = threadIdx.x;           // 0..31
    const int ty = threadIdx.y;           // 0..7
#pragma unroll
    for (int k = 0; k < 32; k += 8) {
        const int b = b0 + ty + k;
        const int i = i0 + tx;
        tile[ty + k][tx] = x[(b * IN_DIM + c) * IN_NUM + i];   // coalesced over i
    }
    __syncthreads();
#pragma unroll
    for (int k = 0; k < 32; k += 8) {
        const int i = i0 + ty + k;
        const int b = b0 + tx;
        xt[(i * IN_DIM + c) * BATCH + b] = tile[tx][ty + k];   // coalesced over b
    }
}

// ---------------------------------------------------------------------------
// u_hat[i][j][b][d] = sum_c W[i][j][d][c] * x_t[i][c][b]
// One wave handles one (i, 16-wide b tile): 10 j-tiles x 2 K-steps of
// v_wmma_f32_16x16x4_f32.  M = d (16), N = b (16), K = c (4 per step).
// ---------------------------------------------------------------------------
__global__ void __launch_bounds__(256)
k_uhat(const float* __restrict__ W, const float* __restrict__ xt,
       float* __restrict__ uhat) {
    const int wave  = threadIdx.x >> 5;
    const int lane  = threadIdx.x & 31;
    const int gw    = blockIdx.x * 8 + wave;     // 18432 waves total
    const int i     = gw >> 4;                   // 0..1151
    const int btile = gw & 15;                   // 0..15
    const int m     = lane & 15;                 // A: M index (=d), B: N index (=b)
    const int half  = lane >> 4;                 // lane half selects K {0,1} vs {2,3}
    const int bb    = btile * 16 + m;

    // B fragments (4x16, fp32): VGPR r, lanes 0-15 -> K=r, lanes 16-31 -> K=r+2
    v2f bfrag[2];
#pragma unroll
    for (int ks = 0; ks < 2; ++ks) {
        const int c0 = ks * 4 + half * 2;
        const float* p = xt + (i * IN_DIM + c0) * BATCH + bb;
        bfrag[ks].x = p[0];
        bfrag[ks].y = p[BATCH];
    }

#pragma unroll
    for (int j = 0; j < OUT_NUM; ++j) {
        // A fragment row: W[i][j][m][.] (contiguous over c)
        const float* wrow = W + ((size_t)((i * OUT_NUM + j) * OUT_DIM + m)) * IN_DIM;
        v8f acc = {};
#pragma unroll
        for (int ks = 0; ks < 2; ++ks) {
            const int c0 = ks * 4 + half * 2;
            v2f afrag;
            afrag.x = wrow[c0];
            afrag.y = wrow[c0 + 1];
            acc = __builtin_amdgcn_wmma_f32_16x16x4_f32(
                false, afrag, false, bfrag[ks], (short)0, acc, false, false);
        }
        // C/D layout: lane holds b=bb fixed, d = 0..7 (lanes 0-15) or 8..15.
        float* outp = uhat + ((size_t)(i * OUT_NUM + j) * BATCH + bb) * OUT_DIM
                           + half * 8;
        ((float4*)outp)[0] = make_float4(acc[0], acc[1], acc[2], acc[3]);
        ((float4*)outp)[1] = make_float4(acc[4], acc[5], acc[6], acc[7]);
    }
}

// ---------------------------------------------------------------------------
__global__ void k_zero(float* __restrict__ p, int n) {
    const int t = blockIdx.x * 256 + threadIdx.x;
    if (t < n) p[t] = 0.0f;
}

// c[i][j] = softmax_j(b_ij[i][j]) ; one thread per i-row
__global__ void k_softmax(const float* __restrict__ bij, float* __restrict__ cij) {
    const int i = blockIdx.x * 256 + threadIdx.x;
    if (i >= IN_NUM) return;
    float r[OUT_NUM];
    float mx = -3.0e38f;
#pragma unroll
    for (int j = 0; j < OUT_NUM; ++j) { r[j] = bij[i * OUT_NUM + j]; mx = fmaxf(mx, r[j]); }
    float s = 0.0f;
#pragma unroll
    for (int j = 0; j < OUT_NUM; ++j) { r[j] = __expf(r[j] - mx); s += r[j]; }
    const float inv = 1.0f / s;
#pragma unroll
    for (int j = 0; j < OUT_NUM; ++j) cij[i * OUT_NUM + j] = r[j] * inv;
}

// ---------------------------------------------------------------------------
// s_part[split][j][b][d] = sum_{i in split} c[i][j] * u_hat[i][j][b][d]
// grid = (ISPLIT, OUT_NUM*16); block thread t covers (b_local, d) -> 256 floats
// of a (j, b-tile) slab; inner loop streams over i with CDNA5 prefetch.
// ---------------------------------------------------------------------------
__global__ void __launch_bounds__(256)
k_reduce_s(const float* __restrict__ uhat, const float* __restrict__ cij,
           float* __restrict__ s_part) {
    const int split = blockIdx.x;
    const int jb    = blockIdx.y;            // j*16 + btile
    const int j     = jb >> 4;
    const int t     = threadIdx.x;
    const int i0    = split * (IN_NUM / ISPLIT);
    const int i1    = i0 + (IN_NUM / ISPLIT);
    const size_t stride = (size_t)OUT_NUM * BATCH * OUT_DIM;      // i stride
    const float* base = uhat + (size_t)jb * 256 + t;              // j*4096 + btile*256 + t
    float acc = 0.0f;
    for (int i = i0; i < i1; ++i) {
        if (i + 8 < i1)   // global_prefetch_b8 the stream 8 i-steps ahead
            __builtin_prefetch(base + (size_t)(i + 8) * stride, 0, 0);
        acc = fmaf(cij[i * OUT_NUM + j], base[(size_t)i * stride], acc);
    }
    s_part[(size_t)split * JBD + jb * 256 + t] = acc;
}

// ---------------------------------------------------------------------------
// v = squash(sum_split s_part); d lives in the low 4 lane bits so the 16-lane
// xor-shuffle reduction stays inside each wave32 half.
// ---------------------------------------------------------------------------
__global__ void __launch_bounds__(256)
k_squash(const float* __restrict__ s_part, float* __restrict__ v) {
    const int flat = blockIdx.x * 256 + threadIdx.x;   // (j*256+b)*16 + d
    float s = 0.0f;
#pragma unroll
    for (int k = 0; k < ISPLIT; ++k) s += s_part[(size_t)k * JBD + flat];
    float sq = s * s;
#pragma unroll
    for (int m = 1; m < 16; m <<= 1) sq += __shfl_xor(sq, m, 32);
    const float scale = sq / ((1.0f + sq) * sqrtf(sq));
    v[flat] = s * scale;
}

// ---------------------------------------------------------------------------
// b_ij[i][j] += (1/B) * sum_{b,d} v[j][b][d] * u_hat[i][j][b][d]
// one block per (i,j): 4096-element dot, wave32 shuffle + LDS tree reduction.
// ---------------------------------------------------------------------------
__global__ void __launch_bounds__(256)
k_update_b(const float* __restrict__ uhat, const float* __restrict__ v,
           float* __restrict__ bij) {
    const int ij = blockIdx.x;               // i*10 + j
    const int j  = ij % OUT_NUM;
    const int t  = threadIdx.x;
    const float* up = uhat + (size_t)ij * (BATCH * OUT_DIM);
    const float* vp = v + (size_t)j * (BATCH * OUT_DIM);
    float acc = 0.0f;
#pragma unroll
    for (int k = 0; k < (BATCH * OUT_DIM) / 256; ++k)
        acc = fmaf(up[k * 256 + t], vp[k * 256 + t], acc);
#pragma unroll
    for (int m = 1; m < 32; m <<= 1) acc += __shfl_xor(acc, m, 32);
    __shared__ float red[8];
    if ((t & 31) == 0) red[t >> 5] = acc;
    __syncthreads();
    if (t == 0) {
        float tot = 0.0f;
#pragma unroll
        for (int w = 0; w < 8; ++w) tot += red[w];
        bij[ij] += tot * (1.0f / (float)BATCH);
    }
}

// out[b][j][d][0] = v[j][b][d]
__global__ void k_out(const float* __restrict__ v, float* __restrict__ out) {
    const int id = blockIdx.x * 256 + threadIdx.x;
    if (id >= BATCH * OUT_NUM * OUT_DIM) return;
    const int d = id & 15;
    const int j = (id >> 4) % OUT_NUM;
    const int b = id / (OUT_NUM * OUT_DIM);
    out[id] = v[((size_t)j * BATCH + b) * OUT_DIM + d];
}

// ---------------------------------------------------------------------------
extern "C" void kernel_launch(void* const* d_in, const int* in_sizes, int n_in,
                              void* d_out, int out_size, void* d_ws, size_t ws_size,
                              hipStream_t stream) {
    const float* x = (const float*)d_in[0];   // [256, 8, 1152]
    const float* W = (const float*)d_in[1];   // [1152, 10, 16, 8]
    float* out = (float*)d_out;               // [256, 10, 16, 1]

    // workspace layout (all offsets 256B-aligned)
    const size_t UHAT_B  = (size_t)IN_NUM * OUT_NUM * BATCH * OUT_DIM * 4; // 188,743,680
    const size_t XT_B    = (size_t)IN_NUM * IN_DIM * BATCH * 4;            //   9,437,184
    const size_t SPART_B = (size_t)ISPLIT * JBD * 4;                       //   1,310,720
    const size_t V_B     = (size_t)JBD * 4;                                //     163,840
    const size_t C_B     = (size_t)IN_NUM * OUT_NUM * 4;                   //      46,080

    char* ws = (char*)d_ws;
    float* uhat   = (float*)(ws);
    float* xt     = (float*)(ws + UHAT_B);
    float* s_part = (float*)(ws + UHAT_B + XT_B);
    float* v      = (float*)(ws + UHAT_B + XT_B + SPART_B);
    float* cij    = (float*)(ws + UHAT_B + XT_B + SPART_B + V_B);
    float* bij    = (float*)(ws + UHAT_B + XT_B + SPART_B + V_B + C_B);
    (void)ws_size; (void)in_sizes; (void)n_in; (void)out_size;

    // 1) retile x for contiguous WMMA B-fragment loads
    k_transpose_x<<<dim3(IN_NUM / 32, BATCH / 32, IN_DIM), dim3(32, 8), 0, stream>>>(x, xt);

    // 2) u_hat via fp32 WMMA: 18432 waves (8 per block)
    k_uhat<<<(IN_NUM * 16) / 8, 256, 0, stream>>>(W, xt, uhat);

    // 3) routing
    k_zero<<<(IN_NUM * OUT_NUM + 255) / 256, 256, 0, stream>>>(bij, IN_NUM * OUT_NUM);
    for (int it = 0; it < NROUTE; ++it) {
        k_softmax<<<(IN_NUM + 255) / 256, 256, 0, stream>>>(bij, cij);
        k_reduce_s<<<dim3(ISPLIT, OUT_NUM * 16), 256, 0, stream>>>(uhat, cij, s_part);
        k_squash<<<JBD / 256, 256, 0, stream>>>(s_part, v);
        if (it + 1 < NROUTE)   // final b-update is dead in the reference
            k_update_b<<<IN_NUM * OUT_NUM, 256, 0, stream>>>(uhat, v, bij);
    }

    // 4) [j,b,d] -> [b,j,d,1]
    k_out<<<(BATCH * OUT_NUM * OUT_DIM) / 256, 256, 0, stream>>>(v, out);
}